// nnChamferDis_35356170781263
// MI455X (gfx1250) — compile-verified
//
#include <hip/hip_runtime.h>
#include <float.h>

typedef __attribute__((ext_vector_type(2))) float v2f;
typedef __attribute__((ext_vector_type(4))) float v4f;
typedef __attribute__((ext_vector_type(8))) float v8f;

#define WPB 8  // waves per block in the main kernel (8 * wave32 = 256 threads)

// v_min3_num_f32 without LLVM's NaN-canonicalize (data is NaN-free by
// construction): one VOP3 op merges two fresh values into the running min.
__device__ __forceinline__ float min3f(float a, float b, float c)
{
    float d;
    asm("v_min3_num_f32 %0, %1, %2, %3" : "=v"(d) : "v"(a), "v"(b), "v"(c));
    return d;
}

// ---------------------------------------------------------------------------
// Prep: transform a point cloud into a paired-tile WMMA B-operand layout.
// B' column j (K=0..3) = (-2x_j, -2y_j, -2z_j, |p_j|^2).
// Lane layout for V_WMMA_F32_16X16X4_F32 B (4x16):
//   lane l, vgpr v  ->  K = 2*(l>>4) + v,  N = l & 15
// Two consecutive 16-column tiles are interleaved so lane l reads BOTH tile
// fragments as one contiguous b128:
//   idx = pair*128 + l*4 + (tile&1)*2 + v
// Padded columns (up to multiple of 32) get (0,0,0,3e38): never win the min.
// ---------------------------------------------------------------------------
__global__ void prep_kernel(const float* __restrict__ pts,
                            float* __restrict__ bp,
                            int npts, int npad)
{
    int j = blockIdx.x * blockDim.x + threadIdx.x;
    if (j >= npad) return;
    float x = 0.f, y = 0.f, z = 0.f, s = 3.0e38f;
    if (j < npts) {
        x = pts[3*j+0]; y = pts[3*j+1]; z = pts[3*j+2];
        s = x*x + y*y + z*z;
    }
    int t = j >> 4, c = j & 15;
    int pair = t >> 1, off = (t & 1) * 2;
    float* b = bp + (size_t)pair * 128 + off;
    b[4*c + 0]        = -2.f * x;   // K=0 -> lane c,    vgpr0
    b[4*c + 1]        = -2.f * y;   // K=1 -> lane c,    vgpr1
    b[4*(16+c) + 0]   = -2.f * z;   // K=2 -> lane c+16, vgpr0
    b[4*(16+c) + 1]   = s;          // K=3 -> lane c+16, vgpr1
}

// ---------------------------------------------------------------------------
// Main: one wave owns a 16-row tile of A, streams all column-tile PAIRS of B
// (pre-swizzled), one V_WMMA_F32_16X16X4_F32 per 16x16 tile, running min via
// v_min3_num_f32 (4 VOP3 per WMMA, the floor). blockIdx.y selects direction.
// Writes one deterministic partial sum per row tile (no float atomics).
// ---------------------------------------------------------------------------
__global__ void __launch_bounds__(32 * WPB)
chamfer_min_kernel(const float* __restrict__ A0, int n0,
                   const float* __restrict__ B0, int np0,
                   float* __restrict__ P0,
                   const float* __restrict__ A1, int n1,
                   const float* __restrict__ B1, int np1,
                   float* __restrict__ P1)
{
    const int dir     = blockIdx.y;
    const float* Ap   = dir ? A1 : A0;
    const float* Bp   = dir ? B1 : B0;
    float*       Pp   = dir ? P1 : P0;
    const int    nA   = dir ? n1 : n0;
    const int    npr  = dir ? np1 : np0;   // number of 32-column tile pairs

    const int lane     = threadIdx.x & 31;
    const int wave     = threadIdx.x >> 5;
    const int rowTile  = blockIdx.x * WPB + wave;
    const int rowTiles = (nA + 15) >> 4;
    if (rowTile >= rowTiles) return;
    const int rowBase  = rowTile << 4;

    // A fragment (16x4, K padded with 1.0):
    //   lanes 0-15 : vgpr0=K0(x), vgpr1=K1(y)   for row M = lane
    //   lanes 16-31: vgpr0=K2(z), vgpr1=K3(1)   for row M = lane-16
    const int  m        = lane & 15;
    const int  hi       = lane >> 4;
    int        arow     = rowBase + m;
    const bool rowValid = (arow < nA);
    if (!rowValid) arow = nA - 1;                  // clamp (excluded from sums)
    const float ax = Ap[3*arow+0], ay = Ap[3*arow+1], az = Ap[3*arow+2];
    const float asq = rowValid ? (ax*ax + ay*ay + az*az) : 0.f;

    v2f afrag;
    afrag.x = hi ? az   : ax;
    afrag.y = hi ? 1.0f : ay;

    v8f rmin;
    #pragma unroll
    for (int i = 0; i < 8; ++i) rmin[i] = FLT_MAX;

    const v8f cz = {0.f,0.f,0.f,0.f,0.f,0.f,0.f,0.f};
    const float* bptr = Bp + 4*lane;

    int p = 0;
    for (; p + 2 <= npr; p += 2) {                 // 2 pairs = 4 tiles / iter
        v4f q0 = *(const v4f*)(bptr + (size_t)(p+0)*128);
        v4f q1 = *(const v4f*)(bptr + (size_t)(p+1)*128);
        v2f b0; b0.x = q0.x; b0.y = q0.y;
        v2f b1; b1.x = q0.z; b1.y = q0.w;
        v2f b2; b2.x = q1.x; b2.y = q1.y;
        v2f b3; b3.x = q1.z; b3.y = q1.w;
        v8f d0 = __builtin_amdgcn_wmma_f32_16x16x4_f32(false, afrag, false, b0, (short)0, cz, false, false);
        v8f d1 = __builtin_amdgcn_wmma_f32_16x16x4_f32(false, afrag, false, b1, (short)0, cz, false, false);
        v8f d2 = __builtin_amdgcn_wmma_f32_16x16x4_f32(false, afrag, false, b2, (short)0, cz, false, false);
        v8f d3 = __builtin_amdgcn_wmma_f32_16x16x4_f32(false, afrag, false, b3, (short)0, cz, false, false);
        #pragma unroll
        for (int i = 0; i < 8; ++i) {
            float t01 = min3f(d0[i], d1[i], d2[i]);
            rmin[i]   = min3f(rmin[i], t01, d3[i]);
        }
    }
    for (; p < npr; ++p) {                         // remainder: 1 pair
        v4f q0 = *(const v4f*)(bptr + (size_t)p*128);
        v2f b0; b0.x = q0.x; b0.y = q0.y;
        v2f b1; b1.x = q0.z; b1.y = q0.w;
        v8f d0 = __builtin_amdgcn_wmma_f32_16x16x4_f32(false, afrag, false, b0, (short)0, cz, false, false);
        v8f d1 = __builtin_amdgcn_wmma_f32_16x16x4_f32(false, afrag, false, b1, (short)0, cz, false, false);
        #pragma unroll
        for (int i = 0; i < 8; ++i)
            rmin[i] = min3f(rmin[i], d0[i], d1[i]);
    }

    // Cross-lane min within each 16-lane half (C/D layout: column N = lane&15).
    #pragma unroll
    for (int msk = 1; msk < 16; msk <<= 1) {
        #pragma unroll
        for (int i = 0; i < 8; ++i)
            rmin[i] = fminf(rmin[i], __shfl_xor(rmin[i], msk, 32));
    }

    // Sum the 16 per-row mins (rows 0-7 in lanes<16, rows 8-15 in lanes>=16).
    float local = 0.f;
    #pragma unroll
    for (int i = 0; i < 8; ++i) {
        int r = rowBase + hi*8 + i;
        if (r < nA) local += rmin[i];
    }
    local += __shfl_xor(local, 16, 32);

    // Add sum of |a_i|^2 over this wave's rows (each row appears in 2 lanes).
    float sq = asq;
    #pragma unroll
    for (int msk = 1; msk < 32; msk <<= 1) sq += __shfl_xor(sq, msk, 32);

    if (lane == 0) Pp[rowTile] = local + 0.5f * sq;
}

// ---------------------------------------------------------------------------
// Deterministic final reduction of the per-row-tile partials -> scalar.
// ---------------------------------------------------------------------------
__global__ void finalize_kernel(const float* __restrict__ p0, int n0,
                                const float* __restrict__ p1, int n1,
                                float* __restrict__ out,
                                float invN, float invM)
{
    __shared__ float sm[256];
    float s0 = 0.f, s1 = 0.f;
    for (int i = threadIdx.x; i < n0; i += 256) s0 += p0[i];
    for (int i = threadIdx.x; i < n1; i += 256) s1 += p1[i];
    sm[threadIdx.x] = s0 * invN + s1 * invM;
    __syncthreads();
    for (int off = 128; off > 0; off >>= 1) {
        if (threadIdx.x < off) sm[threadIdx.x] += sm[threadIdx.x + off];
        __syncthreads();
    }
    if (threadIdx.x == 0) out[0] = sm[0];
}

// ---------------------------------------------------------------------------
extern "C" void kernel_launch(void* const* d_in, const int* in_sizes, int n_in,
                              void* d_out, int out_size, void* d_ws, size_t ws_size,
                              hipStream_t stream)
{
    const float* pc0 = (const float*)d_in[0];
    const float* pc1 = (const float*)d_in[1];
    const int N  = in_sizes[0] / 3;
    const int M  = in_sizes[1] / 3;
    const int Np = (N + 31) & ~31;     // pad to 32 columns (tile pairs)
    const int Mp = (M + 31) & ~31;
    const int rt0 = ((N + 15) >> 4);   // row tiles, direction 0 (pc0 rows)
    const int rt1 = ((M + 15) >> 4);   // row tiles, direction 1 (pc1 rows)

    // Workspace layout (floats):
    //   Bp0: Np*4  (prep of pc0; B operand for direction 1)
    //   Bp1: Mp*4  (prep of pc1; B operand for direction 0)
    //   P0 : rt0   per-row-tile partials, direction 0
    //   P1 : rt1   per-row-tile partials, direction 1
    float* Bp0 = (float*)d_ws;
    float* Bp1 = Bp0 + (size_t)Np * 4;
    float* P0  = Bp1 + (size_t)Mp * 4;
    float* P1  = P0 + rt0;

    prep_kernel<<<(Np + 255) / 256, 256, 0, stream>>>(pc0, Bp0, N, Np);
    prep_kernel<<<(Mp + 255) / 256, 256, 0, stream>>>(pc1, Bp1, M, Mp);

    const int maxRT = rt0 > rt1 ? rt0 : rt1;
    dim3 grid((maxRT + WPB - 1) / WPB, 2);
    chamfer_min_kernel<<<grid, 32 * WPB, 0, stream>>>(
        pc0, N, Bp1, Mp >> 5, P0,      // dir0: A=pc0, B=prep(pc1), tile pairs
        pc1, M, Bp0, Np >> 5, P1);     // dir1: A=pc1, B=prep(pc0)

    finalize_kernel<<<1, 256, 0, stream>>>(P0, rt0, P1, rt1, (float*)d_out,
                                           1.0f / (float)N, 1.0f / (float)M);
}